// EstimatorQNN_65481071400810
// MI455X (gfx1250) — compile-verified
//
#include <hip/hip_runtime.h>
#include <hip/hip_bf16.h>
#include <math.h>

// ---------------------------------------------------------------------------
// CDNA5 fp32 WMMA: V_WMMA_F32_16X16X4_F32
//   A: 16x4 f32  -> 2 VGPRs/lane (lane l<16: row l, K=2g..2g+1 per lane-group g)
//   B: 4x16 f32  -> 2 VGPRs/lane (lane n: col n, K=2g..2g+1 per lane-group g)
//   C/D: 16x16   -> 8 VGPRs (VGPR r, lane n, group g -> M=r+8g, N=n)
// ---------------------------------------------------------------------------
typedef __attribute__((ext_vector_type(2))) float v2f;
typedef __attribute__((ext_vector_type(4))) float f4;
typedef __attribute__((ext_vector_type(8))) float v8f;

__device__ __forceinline__ v8f wmma_f32(v2f a, v2f b, v8f c) {
    // 8 args: (neg_a, A, neg_b, B, c_mod, C, reuse_a, reuse_b)
    return __builtin_amdgcn_wmma_f32_16x16x4_f32(false, a, false, b,
                                                 (short)0, c, false, false);
}

// LDS tile stride: 68 floats = 272B  (16B aligned for float4, +4 banks/row)
#define LDT 68

// ---------------------------------------------------------------------------
// Generic fp32 WMMA GEMM:  out[M,N] = act(A[M,K] @ W[N,K]^T + bias[N])
// TANH: apply tanhf.  NORM: N must be 64; also emit normed = out/(||row||+eps)
// Block = 128 threads (4 waves). Tile = 64x64, K streamed in chunks of 64.
// ---------------------------------------------------------------------------
template<bool TANH, bool NORM>
__global__ __launch_bounds__(128)
void mlp_gemm(const float* __restrict__ A, const float* __restrict__ W,
              const float* __restrict__ bias, float* __restrict__ out,
              float* __restrict__ normed, int M, int N, int K)
{
    __shared__ float As[64 * LDT];
    __shared__ float Ws[64 * LDT];
    __shared__ float Os[NORM ? (64 * LDT) : 1];
    __shared__ float normS[NORM ? 64 : 1];

    const int t    = threadIdx.x;
    const int wave = t >> 5;
    const int lane = t & 31;
    const int l    = lane & 15;   // row/col within 16
    const int g    = lane >> 4;   // lane-group: selects K pair / M half

    const int mtile = blockIdx.x * 64;
    const int ntile = blockIdx.y * 64;

    v8f acc[4] = {};  // wave owns rows [wave*16..+15], 4 N-subtiles of 16

    for (int kk = 0; kk < K; kk += 64) {
        __syncthreads();
        // cooperative tile load: 64 rows x 16 float4 each for A and W
        #pragma unroll
        for (int i = 0; i < 8; ++i) {
            int idx = t + i * 128;        // 0..1023
            int row = idx >> 4;
            int q   = idx & 15;
            f4 va = *(const f4*)(A + (size_t)(mtile + row) * K + kk + q * 4);
            *(f4*)&As[row * LDT + q * 4] = va;
            f4 vw = *(const f4*)(W + (size_t)(ntile + row) * K + kk + q * 4);
            *(f4*)&Ws[row * LDT + q * 4] = vw;
        }
        __syncthreads();

        const float* ar = &As[(wave * 16 + l) * LDT];
        #pragma unroll
        for (int ks = 0; ks < 16; ++ks) {       // 16 x (K=4) steps
            const int kb = ks * 4 + 2 * g;
            v2f a; a.x = ar[kb]; a.y = ar[kb + 1];
            #pragma unroll
            for (int c = 0; c < 4; ++c) {
                const float* br = &Ws[(c * 16 + l) * LDT];
                v2f b; b.x = br[kb]; b.y = br[kb + 1];
                acc[c] = wmma_f32(a, b, acc[c]);
            }
        }
    }

    // epilogue: bias + activation
    #pragma unroll
    for (int c = 0; c < 4; ++c) {
        float bv = bias[ntile + c * 16 + l];
        #pragma unroll
        for (int r = 0; r < 8; ++r) {
            float v = acc[c][r] + bv;
            if (TANH) v = tanhf(v);
            acc[c][r] = v;
        }
    }

    if (!NORM) {
        #pragma unroll
        for (int c = 0; c < 4; ++c)
            #pragma unroll
            for (int r = 0; r < 8; ++r) {
                int row = mtile + wave * 16 + r + 8 * g;
                int col = ntile + c * 16 + l;
                out[(size_t)row * N + col] = acc[c][r];
            }
    } else {
        // N == 64: block owns full rows -> fuse row-norm + normalize
        __syncthreads();
        #pragma unroll
        for (int c = 0; c < 4; ++c)
            #pragma unroll
            for (int r = 0; r < 8; ++r)
                Os[(wave * 16 + r + 8 * g) * LDT + c * 16 + l] = acc[c][r];
        __syncthreads();
        if (t < 64) {
            float s = 0.f;
            #pragma unroll 8
            for (int j = 0; j < 64; ++j) { float v = Os[t * LDT + j]; s += v * v; }
            normS[t] = sqrtf(s) + 1e-12f;   // norm + EPS (reference)
        }
        __syncthreads();
        #pragma unroll
        for (int i = 0; i < 32; ++i) {
            int idx = t + i * 128;        // 0..4095
            int row = idx >> 6;
            int col = idx & 63;
            float v = Os[row * LDT + col];
            out   [(size_t)(mtile + row) * 64 + col] = v;
            normed[(size_t)(mtile + row) * 64 + col] = v / normS[row];
        }
    }
}

// ---------------------------------------------------------------------------
// Fused adjacency:  res = adj @ out, adj_ij = ((n_i . n_j)^2 >= 0.9) & (i!=j)
// Never materializes the 8192x8192 matrix. Block owns 64 rows; streams 64-row
// j-tiles of (normed, out) through LDS. Both GEMMs run on fp32 WMMA.
// ---------------------------------------------------------------------------
__global__ __launch_bounds__(128)
void adj_matmul(const float* __restrict__ outm, const float* __restrict__ normed,
                float* __restrict__ res, int Bsz)
{
    __shared__ float Ni[64 * LDT];   // resident i-tile of normed
    __shared__ float Nj[64 * LDT];   // j-tile of normed; reused as adjacency buf
    __shared__ float Oj[64 * LDT];   // j-tile of out

    const int t    = threadIdx.x;
    const int wave = t >> 5;
    const int lane = t & 31;
    const int l    = lane & 15;
    const int g    = lane >> 4;

    const int mtile = blockIdx.x * 64;

    #pragma unroll
    for (int i = 0; i < 8; ++i) {
        int idx = t + i * 128; int row = idx >> 4; int q = idx & 15;
        *(f4*)&Ni[row * LDT + q * 4] =
            *(const f4*)(normed + (size_t)(mtile + row) * 64 + q * 4);
    }

    v8f acc[4] = {};   // wave's 16 rows x 64 output features

    for (int jt = 0; jt < Bsz; jt += 64) {
        __syncthreads();               // previous iter done with Nj/Oj
        #pragma unroll
        for (int i = 0; i < 8; ++i) {
            int idx = t + i * 128; int row = idx >> 4; int q = idx & 15;
            *(f4*)&Nj[row * LDT + q * 4] =
                *(const f4*)(normed + (size_t)(jt + row) * 64 + q * 4);
            *(f4*)&Oj[row * LDT + q * 4] =
                *(const f4*)(outm + (size_t)(jt + row) * 64 + q * 4);
        }
        if (jt + 64 < Bsz) {           // global_prefetch_b8 next j-tile
            __builtin_prefetch(normed + (size_t)(jt + 64 + (t >> 1)) * 64, 0, 0);
            __builtin_prefetch(outm   + (size_t)(jt + 64 + (t >> 1)) * 64, 0, 0);
        }
        __syncthreads();

        // S = Ni(wave rows) @ Nj^T : 16x64, K = 64
        v8f S[4] = {};
        const float* ar = &Ni[(wave * 16 + l) * LDT];
        #pragma unroll
        for (int ks = 0; ks < 16; ++ks) {
            const int kb = ks * 4 + 2 * g;
            v2f a; a.x = ar[kb]; a.y = ar[kb + 1];
            #pragma unroll
            for (int c = 0; c < 4; ++c) {
                const float* br = &Nj[(c * 16 + l) * LDT];
                v2f b; b.x = br[kb]; b.y = br[kb + 1];
                S[c] = wmma_f32(a, b, S[c]);
            }
        }

        // threshold in registers: fid = S^2 >= 0.9, no self-edges
        #pragma unroll
        for (int c = 0; c < 4; ++c)
            #pragma unroll
            for (int r = 0; r < 8; ++r) {
                int gi = mtile + wave * 16 + r + 8 * g;
                int gj = jt + c * 16 + l;
                float fv = S[c][r]; fv *= fv;
                S[c][r] = (fv >= 0.9f && gi != gj) ? 1.0f : 0.0f;
            }

        __syncthreads();               // all waves done reading Nj as data
        // stage adjacency into Nj's space (disjoint 16-row region per wave):
        // converts D-layout regs -> A-layout reads (same wave, LDS in-order)
        #pragma unroll
        for (int c = 0; c < 4; ++c)
            #pragma unroll
            for (int r = 0; r < 8; ++r)
                Nj[(wave * 16 + r + 8 * g) * LDT + c * 16 + l] = S[c][r];

        // acc += adj(16x64) @ Oj(64x64), K = 64
        const float* a2 = &Nj[(wave * 16 + l) * LDT];
        #pragma unroll
        for (int ks = 0; ks < 16; ++ks) {
            const int kb = ks * 4 + 2 * g;
            v2f a; a.x = a2[kb]; a.y = a2[kb + 1];
            #pragma unroll
            for (int c = 0; c < 4; ++c) {
                v2f b;
                b.x = Oj[kb * LDT + c * 16 + l];
                b.y = Oj[(kb + 1) * LDT + c * 16 + l];
                acc[c] = wmma_f32(a, b, acc[c]);
            }
        }
    }

    #pragma unroll
    for (int c = 0; c < 4; ++c)
        #pragma unroll
        for (int r = 0; r < 8; ++r) {
            int row = mtile + wave * 16 + r + 8 * g;
            res[(size_t)row * 64 + c * 16 + l] = acc[c][r];
        }
}

// ---------------------------------------------------------------------------
extern "C" void kernel_launch(void* const* d_in, const int* in_sizes, int n_in,
                              void* d_out, int out_size, void* d_ws, size_t ws_size,
                              hipStream_t stream) {
    (void)in_sizes; (void)n_in; (void)out_size; (void)ws_size;
    const float* x  = (const float*)d_in[0];
    const float* W1 = (const float*)d_in[1];
    const float* b1 = (const float*)d_in[2];
    const float* W2 = (const float*)d_in[3];
    const float* b2 = (const float*)d_in[4];
    const float* W3 = (const float*)d_in[5];
    const float* b3 = (const float*)d_in[6];
    float* out = (float*)d_out;

    const int B = 8192, D_in = 256, H1 = 512, H2 = 256, D_out = 64;

    char* ws = (char*)d_ws;
    float* h1 = (float*)ws;                                    // B*H1
    float* h2 = (float*)(ws + (size_t)B * H1 * 4);             // B*H2
    float* o  = (float*)(ws + (size_t)B * (H1 + H2) * 4);      // B*D_out
    float* nd = o + (size_t)B * D_out;                         // B*D_out

    dim3 blk(128);
    // layer 1: [8192,256] @ [512,256]^T -> tanh -> h1
    mlp_gemm<true,  false><<<dim3(B / 64, H1 / 64), blk, 0, stream>>>(
        x, W1, b1, h1, nullptr, B, H1, D_in);
    // layer 2: [8192,512] @ [256,512]^T -> tanh -> h2
    mlp_gemm<true,  false><<<dim3(B / 64, H2 / 64), blk, 0, stream>>>(
        h1, W2, b2, h2, nullptr, B, H2, H1);
    // layer 3 + row norms: out, normed
    mlp_gemm<false, true ><<<dim3(B / 64, 1), blk, 0, stream>>>(
        h2, W3, b3, o, nd, B, D_out, H2);
    // fused fidelity-threshold-adjacency matmul
    adj_matmul<<<dim3(B / 64), blk, 0, stream>>>(o, nd, out, B);
}